// PR_1984274890967
// MI455X (gfx1250) — compile-verified
//
#include <hip/hip_runtime.h>
#include <math.h>

#define B_     64
#define CIN_   32
#define COUT_  32
#define MODES_ 16
#define T_     1024
#define DT_    0.01f
#define TWO_PI_ 6.28318530717958647692f

typedef __attribute__((ext_vector_type(2))) float v2f;
typedef __attribute__((ext_vector_type(8))) float v8f;

// D = A(16x4,f32) * B(4x16,f32) + C(16x16,f32)
static __device__ inline v8f wmma_f32(v2f a, v2f b, v8f c) {
    return __builtin_amdgcn_wmma_f32_16x16x4_f32(false, a, false, b, (short)0, c,
                                                 false, false);
}

// ---------------------------------------------------------------------------
// Kernel 1: DFT via WMMA.  alphaT[x][b*32+i] = sum_z x[b,i,z] * e^{-2pi i x z/T}
// rows (M) = flattened (b,i) = 2048, cols (N) = freq = 1024, K = z = 1024.
// Twiddles generated on the fly with exact mod-1024 phase.
// ---------------------------------------------------------------------------
__global__ void __launch_bounds__(256) k_dft(const float* __restrict__ x,
                                             float* __restrict__ aR,
                                             float* __restrict__ aI) {
    const int wave = threadIdx.x >> 5;
    const int lane = threadIdx.x & 31;
    const int ln   = lane & 15;
    const int hf   = lane >> 4;   // 0|1
    const int k0   = hf * 2;      // K sub-offset {0,2}
    const int tile = blockIdx.x * 8 + wave;   // 0..8191
    const int tm = tile >> 6;                 // 0..127 row tile (bi)
    const int tn = tile & 63;                 // 0..63  freq tile
    const int m0 = tm * 16;
    const int f  = tn * 16 + ln;              // lane's frequency (B col)
    const float w = TWO_PI_ / (float)T_;

    v8f accC = {0,0,0,0,0,0,0,0};
    v8f accS = {0,0,0,0,0,0,0,0};
    const float* xrow = x + (size_t)(m0 + ln) * T_;   // A row (bi = m0+ln)
    for (int kk = 0; kk < T_ / 4; ++kk) {
        const int z0 = kk * 4 + k0;
        v2f a; a.x = xrow[z0]; a.y = xrow[z0 + 1];
        float s0, c0, s1, c1;
        __sincosf(w * (float)((f * z0) & (T_ - 1)), &s0, &c0);
        __sincosf(w * (float)((f * (z0 + 1)) & (T_ - 1)), &s1, &c1);
        v2f bc; bc.x = c0; bc.y = c1;
        v2f bs; bs.x = s0; bs.y = s1;
        accC = wmma_f32(a, bc, accC);
        accS = wmma_f32(a, bs, accS);
    }
#pragma unroll
    for (int v = 0; v < 8; ++v) {
        const int bi = m0 + v + hf * 8;
        aR[(size_t)f * (B_ * CIN_) + bi] =  accC[v];
        aI[(size_t)f * (B_ * CIN_) + bi] = -accS[v];
    }
}

// ---------------------------------------------------------------------------
// Kernel 2: H[x][i*32+o] = sum_m residue/(i*lam[x] - pole), pure VALU.
// ---------------------------------------------------------------------------
__global__ void __launch_bounds__(256) k_hsum(const float* __restrict__ pr,
                                              const float* __restrict__ pim,
                                              const float* __restrict__ resr,
                                              const float* __restrict__ resi,
                                              float* __restrict__ HR,
                                              float* __restrict__ HI) {
    const int idx = blockIdx.x * 256 + threadIdx.x;  // 0..(T*CIN*COUT-1)
    const int xf = idx >> 10;
    const int r  = idx & 1023;
    const int i  = r >> 5;
    const int o  = r & 31;
    const int fsh = (xf < T_ / 2) ? xf : xf - T_;
    const float lamI = (TWO_PI_ / ((float)T_ * DT_)) * (float)fsh;
    float sr = 0.f, si = 0.f;
#pragma unroll
    for (int m = 0; m < MODES_; ++m) {
        const int p = (i * COUT_ + o) * MODES_ + m;
        const float dr = -pr[p];
        const float di = lamI - pim[p];
        const float inv = 1.0f / (dr * dr + di * di);
        const float R = resr[p], I = resi[p];
        sr += (R * dr + I * di) * inv;
        si += (I * dr - R * di) * inv;
    }
    HR[idx] = sr;
    HI[idx] = si;
}

// ---------------------------------------------------------------------------
// Kernel 3: per-frequency complex GEMM  out1T[x][b*32+o] = sum_i alpha*H.
// One block per frequency; 8 waves cover the 4x2 tile grid of a 64x32 output.
// ---------------------------------------------------------------------------
__global__ void __launch_bounds__(256) k_out1(const float* __restrict__ aR,
                                              const float* __restrict__ aI,
                                              const float* __restrict__ HR,
                                              const float* __restrict__ HI,
                                              float* __restrict__ o1R,
                                              float* __restrict__ o1I) {
    const int wave = threadIdx.x >> 5;
    const int lane = threadIdx.x & 31;
    const int ln = lane & 15;
    const int hf = lane >> 4;
    const int k0 = hf * 2;
    const int xf = blockIdx.x;
    const int tm = wave >> 1;       // 0..3  (b tile)
    const int tn = wave & 1;        // 0..1  (o tile)
    const float* aRx = aR + (size_t)xf * (B_ * CIN_);
    const float* aIx = aI + (size_t)xf * (B_ * CIN_);
    const float* HRx = HR + (size_t)xf * (CIN_ * COUT_);
    const float* HIx = HI + (size_t)xf * (CIN_ * COUT_);
    const int brow = tm * 16 + ln;
    const int ocol = tn * 16 + ln;

    v8f RR = {0,0,0,0,0,0,0,0}, II = RR, RI = RR, IR = RR;
#pragma unroll
    for (int kk = 0; kk < CIN_ / 4; ++kk) {
        const int i0 = kk * 4 + k0;
        v2f ar; ar.x = aRx[brow * CIN_ + i0]; ar.y = aRx[brow * CIN_ + i0 + 1];
        v2f ai; ai.x = aIx[brow * CIN_ + i0]; ai.y = aIx[brow * CIN_ + i0 + 1];
        v2f hr; hr.x = HRx[i0 * COUT_ + ocol]; hr.y = HRx[(i0 + 1) * COUT_ + ocol];
        v2f hi; hi.x = HIx[i0 * COUT_ + ocol]; hi.y = HIx[(i0 + 1) * COUT_ + ocol];
        RR = wmma_f32(ar, hr, RR);
        II = wmma_f32(ai, hi, II);
        RI = wmma_f32(ar, hi, RI);
        IR = wmma_f32(ai, hr, IR);
    }
#pragma unroll
    for (int v = 0; v < 8; ++v) {
        const int b = tm * 16 + v + hf * 8;
        const int o = tn * 16 + ln;
        o1R[(size_t)xf * (B_ * COUT_) + b * COUT_ + o] = RR[v] - II[v];
        o1I[(size_t)xf * (B_ * COUT_) + b * COUT_ + o] = RI[v] + IR[v];
    }
}

// ---------------------------------------------------------------------------
// Kernel 4: IDFT via WMMA, writes x1 directly to d_out.
// x1[bo][z] = (1/T) * sum_x ( o1R[x][bo]*cos(2pi x z/T) - o1I[x][bo]*sin )
// ---------------------------------------------------------------------------
__global__ void __launch_bounds__(256) k_ifft(const float* __restrict__ o1R,
                                              const float* __restrict__ o1I,
                                              float* __restrict__ out) {
    const int wave = threadIdx.x >> 5;
    const int lane = threadIdx.x & 31;
    const int ln = lane & 15;
    const int hf = lane >> 4;
    const int k0 = hf * 2;
    const int tile = blockIdx.x * 8 + wave;  // 0..8191
    const int tm = tile >> 6;                // 0..127 (bo rows)
    const int tn = tile & 63;                // 0..63  (z cols)
    const int m0 = tm * 16;
    const int zc = tn * 16 + ln;
    const float w = TWO_PI_ / (float)T_;

    v8f accC = {0,0,0,0,0,0,0,0};
    v8f accS = {0,0,0,0,0,0,0,0};
    const int borow = m0 + ln;
    for (int kk = 0; kk < T_ / 4; ++kk) {
        const int x0 = kk * 4 + k0;
        v2f ar; ar.x = o1R[(size_t)x0 * (B_ * COUT_) + borow];
        ar.y = o1R[(size_t)(x0 + 1) * (B_ * COUT_) + borow];
        v2f ai; ai.x = o1I[(size_t)x0 * (B_ * COUT_) + borow];
        ai.y = o1I[(size_t)(x0 + 1) * (B_ * COUT_) + borow];
        float s0, c0, s1, c1;
        __sincosf(w * (float)((x0 * zc) & (T_ - 1)), &s0, &c0);
        __sincosf(w * (float)(((x0 + 1) * zc) & (T_ - 1)), &s1, &c1);
        v2f bc; bc.x = c0; bc.y = c1;
        v2f bs; bs.x = s0; bs.y = s1;
        accC = wmma_f32(ar, bc, accC);
        accS = wmma_f32(ai, bs, accS);
    }
#pragma unroll
    for (int v = 0; v < 8; ++v) {
        const int bo = m0 + v + hf * 8;
        out[(size_t)bo * T_ + tn * 16 + ln] = (accC[v] - accS[v]) * (1.0f / (float)T_);
    }
}

// ---------------------------------------------------------------------------
// Kernel 5: out2 partials.  out2[b][o*16+k] = -sum_{x,i} alpha[x,b,i]*Hw[x,i,o,k]
// K = T*CIN = 32768 split into 16 chunks (deterministic partial sums).
// Hw generated on the fly: one complex reciprocal per element.
// ---------------------------------------------------------------------------
__global__ void __launch_bounds__(256) k_out2(const float* __restrict__ aR,
                                              const float* __restrict__ aI,
                                              const float* __restrict__ pr,
                                              const float* __restrict__ pim,
                                              const float* __restrict__ resr,
                                              const float* __restrict__ resi,
                                              float* __restrict__ pRbuf,
                                              float* __restrict__ pIbuf) {
    const int wave = threadIdx.x >> 5;
    const int lane = threadIdx.x & 31;
    const int ln = lane & 15;
    const int hf = lane >> 4;
    const int k0 = hf * 2;
    const int tile = (blockIdx.x & 15) * 8 + wave;  // 0..127
    const int kc   = blockIdx.x >> 4;               // 0..15 K-chunk
    const int tm = tile >> 5;   // 0..3  (b tile)
    const int tn = tile & 31;   // 0..31 ((o,k) tile)
    const int m0 = tm * 16;
    const int ncol = tn * 16 + ln;
    const int oo = ncol >> 4;
    const int mm = ncol & 15;
    const int brow = m0 + ln;

    v8f RR = {0,0,0,0,0,0,0,0}, II = RR, RI = RR, IR = RR;
    const int kk0 = kc * 512;
    for (int kk = kk0; kk < kk0 + 512; ++kk) {
        const int kap = kk * 4;
        const int xf = kap >> 5;                 // frequency
        const int i0 = (kap & 31) + k0;          // in-channel
        const float* aRx = aR + (size_t)xf * (B_ * CIN_) + brow * CIN_;
        const float* aIx = aI + (size_t)xf * (B_ * CIN_) + brow * CIN_;
        v2f ar; ar.x = aRx[i0]; ar.y = aRx[i0 + 1];
        v2f ai; ai.x = aIx[i0]; ai.y = aIx[i0 + 1];
        const int fsh = (xf < T_ / 2) ? xf : xf - T_;
        const float lamI = (TWO_PI_ / ((float)T_ * DT_)) * (float)fsh;
        v2f wr, wi;
        {
            int p = (i0 * COUT_ + oo) * MODES_ + mm;
            float dr = -pr[p], di = lamI - pim[p];
            float inv = 1.0f / (dr * dr + di * di);
            float R = resr[p], I = resi[p];
            wr.x = (R * dr + I * di) * inv;
            wi.x = (I * dr - R * di) * inv;
            p += COUT_ * MODES_;   // i0 + 1
            dr = -pr[p]; di = lamI - pim[p];
            inv = 1.0f / (dr * dr + di * di);
            R = resr[p]; I = resi[p];
            wr.y = (R * dr + I * di) * inv;
            wi.y = (I * dr - R * di) * inv;
        }
        RR = wmma_f32(ar, wr, RR);
        II = wmma_f32(ai, wi, II);
        RI = wmma_f32(ar, wi, RI);
        IR = wmma_f32(ai, wr, IR);
    }
#pragma unroll
    for (int v = 0; v < 8; ++v) {
        const int b = m0 + v + hf * 8;
        const int n = tn * 16 + ln;
        // out2 = -(alpha*Hw):  real = II-RR, imag = -(RI+IR)
        pRbuf[(size_t)kc * (B_ * COUT_ * MODES_) + b * 512 + n] = II[v] - RR[v];
        pIbuf[(size_t)kc * (B_ * COUT_ * MODES_) + b * 512 + n] = -(RI[v] + IR[v]);
    }
}

__global__ void __launch_bounds__(256) k_red(const float* __restrict__ pRbuf,
                                             const float* __restrict__ pIbuf,
                                             float* __restrict__ o2R,
                                             float* __restrict__ o2I) {
    const int idx = blockIdx.x * 256 + threadIdx.x;  // 0..32767
    float sR = 0.f, sI = 0.f;
#pragma unroll
    for (int c = 0; c < 16; ++c) {
        sR += pRbuf[(size_t)c * (B_ * COUT_ * MODES_) + idx];
        sI += pIbuf[(size_t)c * (B_ * COUT_ * MODES_) + idx];
    }
    o2R[idx] = sR;
    o2I[idx] = sI;
}

// ---------------------------------------------------------------------------
// Kernel 6: x2[b,o,z] = (1/T) Re( sum_{c,m} out2[b,c,m] * e^{pole[c,o,m]*t[z]} )
// added into d_out.  K = CIN*MODES = 512; weights via exp+sincos on the fly.
// ---------------------------------------------------------------------------
__global__ void __launch_bounds__(256) k_x2(const float* __restrict__ o2R,
                                            const float* __restrict__ o2I,
                                            const float* __restrict__ pr,
                                            const float* __restrict__ pim,
                                            const float* __restrict__ tg,
                                            float* __restrict__ out) {
    const int wave = threadIdx.x >> 5;
    const int lane = threadIdx.x & 31;
    const int ln = lane & 15;
    const int hf = lane >> 4;
    const int k0 = hf * 2;
    const int tile = blockIdx.x * 8 + wave;  // 0..8191
    const int tm = tile & 3;                 // b tile
    const int rest = tile >> 2;              // 0..2047
    const int oo = rest >> 6;                // 0..31 out channel
    const int zt = rest & 63;                // z tile
    const int m0 = tm * 16;
    const int z = zt * 16 + ln;
    const float tz = tg[z];
    const int brow = m0 + ln;

    v8f A1 = {0,0,0,0,0,0,0,0}, A2 = A1;
    for (int kk = 0; kk < (CIN_ * MODES_) / 4; ++kk) {
        const int kap = kk * 4;
        const int c = kap >> 4;
        const int mb = (kap & 15) + k0;
        v2f ar; ar.x = o2R[brow * 512 + kap + k0]; ar.y = o2R[brow * 512 + kap + k0 + 1];
        v2f ai; ai.x = o2I[brow * 512 + kap + k0]; ai.y = o2I[brow * 512 + kap + k0 + 1];
        v2f wr, wi;
        {
            int p = (c * COUT_ + oo) * MODES_ + mb;
            float e = __expf(pr[p] * tz);
            float s, co; __sincosf(pim[p] * tz, &s, &co);
            wr.x = e * co; wi.x = e * s;
            p += 1;
            e = __expf(pr[p] * tz);
            __sincosf(pim[p] * tz, &s, &co);
            wr.y = e * co; wi.y = e * s;
        }
        A1 = wmma_f32(ar, wr, A1);
        A2 = wmma_f32(ai, wi, A2);
    }
#pragma unroll
    for (int v = 0; v < 8; ++v) {
        const int b = m0 + v + hf * 8;
        const size_t oidx = ((size_t)(b * COUT_ + oo)) * T_ + zt * 16 + ln;
        out[oidx] += (A1[v] - A2[v]) * (1.0f / (float)T_);
    }
}

// ---------------------------------------------------------------------------
extern "C" void kernel_launch(void* const* d_in, const int* in_sizes, int n_in,
                              void* d_out, int out_size, void* d_ws, size_t ws_size,
                              hipStream_t stream) {
    (void)in_sizes; (void)n_in; (void)out_size; (void)ws_size;
    const float* x    = (const float*)d_in[0];
    const float* tg   = (const float*)d_in[1];
    const float* pr   = (const float*)d_in[2];
    const float* pim  = (const float*)d_in[3];
    const float* resr = (const float*)d_in[4];
    const float* resi = (const float*)d_in[5];
    float* out = (float*)d_out;

    const size_t nA  = (size_t)T_ * B_ * CIN_;        // 2,097,152
    const size_t nH  = (size_t)T_ * CIN_ * COUT_;     // 1,048,576
    const size_t nO1 = (size_t)T_ * B_ * COUT_;       // 2,097,152
    const size_t nO2 = (size_t)B_ * COUT_ * MODES_;   //    32,768
    float* ws  = (float*)d_ws;
    float* aR  = ws;
    float* aI  = aR + nA;
    float* HR  = aI + nA;
    float* HI  = HR + nH;
    float* o1R = HI + nH;
    float* o1I = o1R + nO1;
    float* o2R = o1I + nO1;
    float* o2I = o2R + nO2;
    float* pRb = o2I + nO2;
    float* pIb = pRb + 16 * nO2;

    k_dft <<<1024, 256, 0, stream>>>(x, aR, aI);
    k_hsum<<<4096, 256, 0, stream>>>(pr, pim, resr, resi, HR, HI);
    k_out1<<<1024, 256, 0, stream>>>(aR, aI, HR, HI, o1R, o1I);
    k_ifft<<<1024, 256, 0, stream>>>(o1R, o1I, out);
    k_out2<<<256, 256, 0, stream>>>(aR, aI, pr, pim, resr, resi, pRb, pIb);
    k_red <<<128, 256, 0, stream>>>(pRb, pIb, o2R, o2I);
    k_x2  <<<1024, 256, 0, stream>>>(o2R, o2I, pr, pim, tg, out);
}